// SVF_3D_34900904247316
// MI455X (gfx1250) — compile-verified
//
#include <hip/hip_runtime.h>

// SVF scaling-and-squaring (3D trilinear grid_sample, border, align_corners)
// for v: (2,3,128,128,128) f32.  Internal displacement buffers use AoS float3
// so each of the 8 corner gathers is a single b96 load; the coalesced per-block
// self-tile is staged into LDS via gfx1250 async global->LDS loads.

#define DIMS   128
#define PLANE  (DIMS * DIMS * DIMS)   // 2^21
#define NBATCH 2
#define VOX    (NBATCH * PLANE)       // 4,194,304 voxels
#define NSTEPS 12
#define BLOCK  256

#if defined(__gfx1250__) && __has_builtin(__builtin_amdgcn_global_load_async_to_lds_b32)
#define HAVE_ASYNC_LDS 1
#else
#define HAVE_ASYNC_LDS 0
#endif

struct F3 { float x, y, z; };

__device__ __forceinline__ F3 lerp3(F3 a, F3 b, float w) {
  F3 r;
  r.x = fmaf(w, b.x - a.x, a.x);
  r.y = fmaf(w, b.y - a.y, a.y);
  r.z = fmaf(w, b.z - a.z, a.z);
  return r;
}

// disp0 = v * (2/(dim-1)) / 2^NO_STEPS, SoA(v) -> AoS float3
__global__ void svf_init(const float* __restrict__ v, float* __restrict__ A) {
  const int i = blockIdx.x * BLOCK + threadIdx.x;           // voxel id
  const int n = i >> 21;
  const int r = i & (PLANE - 1);
  const float k = (2.0f / 127.0f) / 4096.0f;
  const float* p = v + (size_t)n * 3 * PLANE + r;
  float* o = A + (size_t)i * 3;
  o[0] = p[0]         * k;
  o[1] = p[PLANE]     * k;
  o[2] = p[2 * PLANE] * k;
}

// disp' = disp + trilinear_sample_border(disp, identity + disp)
__global__ void svf_step(const float* __restrict__ src, float* __restrict__ dst) {
  __shared__ float tile[BLOCK * 3];
  const int t = threadIdx.x;
  const int i = blockIdx.x * BLOCK + t;
  const float* tsrc = src + (size_t)blockIdx.x * (BLOCK * 3);

#if HAVE_ASYNC_LDS
  // Stage this block's 256 float3 self-displacements into LDS with the
  // gfx1250 async DMA path (ASYNCcnt), 3 coalesced b32 issues per lane.
  __builtin_amdgcn_global_load_async_to_lds_b32(
      (__attribute__((address_space(1))) int*)(tsrc + t),
      (__attribute__((address_space(3))) int*)(tile + t), 0, 0);
  __builtin_amdgcn_global_load_async_to_lds_b32(
      (__attribute__((address_space(1))) int*)(tsrc + t + BLOCK),
      (__attribute__((address_space(3))) int*)(tile + t + BLOCK), 0, 0);
  __builtin_amdgcn_global_load_async_to_lds_b32(
      (__attribute__((address_space(1))) int*)(tsrc + t + 2 * BLOCK),
      (__attribute__((address_space(3))) int*)(tile + t + 2 * BLOCK), 0, 0);
  asm volatile("s_wait_asynccnt 0" ::: "memory");
#else
  tile[t]             = tsrc[t];
  tile[t + BLOCK]     = tsrc[t + BLOCK];
  tile[t + 2 * BLOCK] = tsrc[t + 2 * BLOCK];
#endif
  __syncthreads();

  F3 d;
  d.x = tile[3 * t + 0];
  d.y = tile[3 * t + 1];
  d.z = tile[3 * t + 2];

  const int n  = i >> 21;
  const int r  = i & (PLANE - 1);
  const int xi = r & 127;
  const int yi = (r >> 7) & 127;
  const int zi = r >> 14;

  // unnormalized sample coords: (idg + disp + 1) * 0.5 * (dim-1) == coord + disp*63.5
  const float gx = fmaf(d.x, 63.5f, (float)xi);
  const float gy = fmaf(d.y, 63.5f, (float)yi);
  const float gz = fmaf(d.z, 63.5f, (float)zi);

  const float fx = floorf(gx), fy = floorf(gy), fz = floorf(gz);
  const float wx = gx - fx,    wy = gy - fy,    wz = gz - fz;   // weights from UNclamped floor

  int x0 = (int)fminf(fmaxf(fx, -1.0f), 127.0f);
  int y0 = (int)fminf(fmaxf(fy, -1.0f), 127.0f);
  int z0 = (int)fminf(fmaxf(fz, -1.0f), 127.0f);
  const int x1 = (x0 + 1 > 127) ? 127 : x0 + 1;  const int x0c = (x0 < 0) ? 0 : x0;
  const int y1 = (y0 + 1 > 127) ? 127 : y0 + 1;  const int y0c = (y0 < 0) ? 0 : y0;
  const int z1 = (z0 + 1 > 127) ? 127 : z0 + 1;  const int z0c = (z0 < 0) ? 0 : z0;

  const F3* nb = (const F3*)src + (size_t)n * PLANE;
#define FETCH(zz, yy, xx) nb[(size_t)((((zz) * DIMS) + (yy)) * DIMS + (xx))]
  const F3 c000 = FETCH(z0c, y0c, x0c), c001 = FETCH(z0c, y0c, x1);
  const F3 c010 = FETCH(z0c, y1,  x0c), c011 = FETCH(z0c, y1,  x1);
  const F3 c100 = FETCH(z1,  y0c, x0c), c101 = FETCH(z1,  y0c, x1);
  const F3 c110 = FETCH(z1,  y1,  x0c), c111 = FETCH(z1,  y1,  x1);
#undef FETCH

  const F3 e00 = lerp3(c000, c001, wx), e01 = lerp3(c010, c011, wx);
  const F3 e10 = lerp3(c100, c101, wx), e11 = lerp3(c110, c111, wx);
  const F3 f0  = lerp3(e00, e01, wy),   f1  = lerp3(e10, e11, wy);
  const F3 s   = lerp3(f0, f1, wz);

  float* o = dst + (size_t)i * 3;
  o[0] = d.x + s.x;
  o[1] = d.y + s.y;
  o[2] = d.z + s.z;
}

// disp_vox (SoA, second output tensor) = disp / scale = disp * 63.5
__global__ void svf_dispvox(const float* __restrict__ A, float* __restrict__ dv) {
  const int i = blockIdx.x * BLOCK + threadIdx.x;
  const int n = i >> 21;
  const int r = i & (PLANE - 1);
  const float* p = A + (size_t)i * 3;
  const size_t b = (size_t)n * 3 * PLANE + r;
  dv[b]             = p[0] * 63.5f;
  dv[b + PLANE]     = p[1] * 63.5f;
  dv[b + 2 * PLANE] = p[2] * 63.5f;
}

// transformation (SoA, first output tensor) = idg + disp = idg + disp_vox*(2/127)
__global__ void svf_trans(const float* __restrict__ dv, float* __restrict__ tr) {
  const int e = blockIdx.x * BLOCK + threadIdx.x;      // over 3*VOX elements
  const int r  = e & (PLANE - 1);
  const int nc = e >> 21;
  const int c  = nc % 3;
  const int coord = (c == 0) ? (r & 127) : (c == 1) ? ((r >> 7) & 127) : (r >> 14);
  const float idg = fmaf((float)coord, 2.0f / 127.0f, -1.0f);
  tr[e] = fmaf(dv[e], 2.0f / 127.0f, idg);
}

extern "C" void kernel_launch(void* const* d_in, const int* in_sizes, int n_in,
                              void* d_out, int out_size, void* d_ws, size_t ws_size,
                              hipStream_t stream) {
  (void)in_sizes; (void)n_in; (void)out_size;
  const float* v   = (const float*)d_in[0];
  float*       out = (float*)d_out;

  const size_t bufFloats = (size_t)VOX * 3;          // 12,582,912
  const size_t bufBytes  = bufFloats * sizeof(float); // ~50.3 MB
  float* ws = (float*)d_ws;

  // Ping-pong displacement buffers (AoS float3). Prefer workspace (both fit in
  // the 192MB L2 together); degrade gracefully onto d_out halves.
  float* A;
  float* B;
  if (ws_size >= 2 * bufBytes)      { A = ws;              B = ws + bufFloats; }
  else if (ws_size >= bufBytes)     { A = ws;              B = out;            }
  else                              { A = out;             B = out + bufFloats; }

  const dim3 blk(BLOCK);
  const dim3 grdV(VOX / BLOCK);          // 16384 blocks
  const dim3 grdE((3 * VOX) / BLOCK);    // 49152 blocks

  svf_init<<<grdV, blk, 0, stream>>>(v, A);

  float* s = A;
  float* d = B;
  for (int k = 0; k < NSTEPS; ++k) {
    svf_step<<<grdV, blk, 0, stream>>>(s, d);
    float* tmp = s; s = d; d = tmp;
  }
  // NSTEPS is even -> final displacement is back in A (== s).

  // Two race-free elementwise passes (safe even when s aliases out half 0):
  svf_dispvox<<<grdV, blk, 0, stream>>>(s, out + (size_t)3 * VOX);
  svf_trans<<<grdE, blk, 0, stream>>>(out + (size_t)3 * VOX, out);
}